// SingleT2FLS_Mamdani_9165460210233
// MI455X (gfx1250) — compile-verified
//
#include <hip/hip_runtime.h>
#include <math.h>

// T2FLS Mamdani with Karnik-Mendel type reduction.
// Membership stage reformulated as f32 GEMM: sum_a (x-m)^2 * w =
//   [x row | x^2 row] (K=32) @ B(32x64) + const[r],
// executed with V_WMMA_F32_16X16X4_F32 (full f32 precision, K chained 8x).
// KM-left / KM-right scans run on disjoint wave pairs (wave32-uniform split).

typedef __attribute__((ext_vector_type(2))) float v2f;
typedef __attribute__((ext_vector_type(8))) float v8f;

// ---- workspace layout (float units) ----
#define WS_BHI   0      // [32][64]  rows 0..15: -2*m*whi (x part); 16..31: whi (x^2 part)
#define WS_BLO   2048
#define WS_CHI   4096   // [64] sum_a m^2*whi
#define WS_CLO   4160
#define WS_B1    4224   // [64] sorted c1
#define WS_B2    4288   // [64] sorted c2
#define WS_P1    4352   // [64] int argsort(c1)
#define WS_P2    4416   // [64] int argsort(c2)

__global__ void t2fls_setup(const float* __restrict__ W,
                            const float* __restrict__ c1,
                            const float* __restrict__ c2,
                            float* __restrict__ ws) {
  const int r = threadIdx.x;
  if (r >= 64) return;
  float* Bhi = ws + WS_BHI;
  float* Blo = ws + WS_BLO;
  float chi = 0.f, clo = 0.f;
  for (int a = 0; a < 16; ++a) {
    float m  = W[r * 16 + a];
    float s1 = fabsf(W[r * 16 + a + 1]);
    float s2 = fabsf(W[r * 16 + a + 2]);
    float sh = fmaxf(s1, s2);
    float sl = fminf(s1, s2);
    float wh = 1.f / (sh * sh);
    float wl = 1.f / (sl * sl);
    Bhi[a * 64 + r]        = -2.f * m * wh;
    Bhi[(16 + a) * 64 + r] = wh;
    Blo[a * 64 + r]        = -2.f * m * wl;
    Blo[(16 + a) * 64 + r] = wl;
    chi = fmaf(m * m, wh, chi);
    clo = fmaf(m * m, wl, clo);
  }
  ws[WS_CHI + r] = chi;
  ws[WS_CLO + r] = clo;
  // stable ascending rank sort (matches jnp.argsort)
  {
    float key = c1[r];
    int rank = 0;
    for (int j = 0; j < 64; ++j) {
      float v = c1[j];
      rank += (v < key) || (v == key && j < r);
    }
    ws[WS_B1 + rank] = key;
    ((int*)ws)[WS_P1 + rank] = r;
  }
  {
    float key = c2[r];
    int rank = 0;
    for (int j = 0; j < 64; ++j) {
      float v = c2[j];
      rank += (v < key) || (v == key && j < r);
    }
    ws[WS_B2 + rank] = key;
    ((int*)ws)[WS_P2 + rank] = r;
  }
}

// Block: 128 threads = 4 waves; each wave computes a 16-sample x 64-rule tile
// with 64 chained v_wmma_f32_16x16x4_f32. Then waves 0-1 run KM-left and
// waves 2-3 run KM-right in parallel (one thread per sample per side).
__launch_bounds__(128)
__global__ void t2fls_main(const float* __restrict__ x,
                           const float* __restrict__ c1,
                           const float* __restrict__ c2,
                           const float* __restrict__ ws,
                           float* __restrict__ out) {
  __shared__ float sUU[64][65];   // padded stride: conflict-free permuted gathers
  __shared__ float sLL[64][65];
  __shared__ float sB1[64], sB2[64], sC1[64], sC2[64];
  __shared__ int   sP1[64], sP2[64];
  __shared__ float sYL[64];

  const int tid  = threadIdx.x;
  const int lane = tid & 31;
  const int wave = tid >> 5;     // 0..3
  const int h    = lane >> 4;    // K-half selector per WMMA A/B layout
  const int col  = lane & 15;
  const int blockBase = (int)blockIdx.x * 64;

  if (tid < 64) {
    sB1[tid] = ws[WS_B1 + tid];
    sB2[tid] = ws[WS_B2 + tid];
    sC1[tid] = c1[tid];
    sC2[tid] = c2[tid];
    sP1[tid] = ((const int*)ws)[WS_P1 + tid];
    sP2[tid] = ((const int*)ws)[WS_P2 + tid];
  }

  // ---------- Phase 1: membership distances via f32 WMMA ----------
  // A (16x4 f32): lanes 0-15 -> rows M=0..15 holding K={0,1}; lanes 16-31 K={2,3}.
  // B (4x16 f32): lanes 0-15 -> cols N, K rows {0,1} in v0/v1; lanes 16-31 K={2,3}.
  const float* xrow = x + (blockBase + wave * 16 + col) * 16;
  const float* Bhi  = ws + WS_BHI;
  const float* Blo  = ws + WS_BLO;

  v8f accH[4] = {}, accL[4] = {};

  #pragma unroll
  for (int kk = 0; kk < 4; ++kk) {
    const int f0 = kk * 4 + 2 * h;          // this lane's K pair within group
    v2f ax, ax2;
    ax.x  = xrow[f0];
    ax.y  = xrow[f0 + 1];
    ax2.x = ax.x * ax.x;
    ax2.y = ax.y * ax.y;
    #pragma unroll
    for (int t = 0; t < 4; ++t) {           // 4 rule tiles of 16
      const int c = t * 16 + col;
      v2f bh, bh2, bl, bl2;
      bh.x  = Bhi[(f0)      * 64 + c];  bh.y  = Bhi[(f0 + 1)  * 64 + c];
      bh2.x = Bhi[(16 + f0) * 64 + c];  bh2.y = Bhi[(17 + f0) * 64 + c];
      bl.x  = Blo[(f0)      * 64 + c];  bl.y  = Blo[(f0 + 1)  * 64 + c];
      bl2.x = Blo[(16 + f0) * 64 + c];  bl2.y = Blo[(17 + f0) * 64 + c];
      accH[t] = __builtin_amdgcn_wmma_f32_16x16x4_f32(false, ax,  false, bh,  (short)0, accH[t], false, false);
      accH[t] = __builtin_amdgcn_wmma_f32_16x16x4_f32(false, ax2, false, bh2, (short)0, accH[t], false, false);
      accL[t] = __builtin_amdgcn_wmma_f32_16x16x4_f32(false, ax,  false, bl,  (short)0, accL[t], false, false);
      accL[t] = __builtin_amdgcn_wmma_f32_16x16x4_f32(false, ax2, false, bl2, (short)0, accL[t], false, false);
    }
  }

  // ---------- Phase 2: add per-rule constant, exp, spill to LDS ----------
  // C layout: VGPR i -> M = i (lanes 0-15) / M = i+8 (lanes 16-31), N = lane&15.
  const float* chi = ws + WS_CHI;
  const float* clo = ws + WS_CLO;
  #pragma unroll
  for (int t = 0; t < 4; ++t) {
    const int r  = t * 16 + col;
    const float ch = chi[r];
    const float cl = clo[r];
    #pragma unroll
    for (int i = 0; i < 8; ++i) {
      const int m = wave * 16 + i + 8 * h;
      sUU[m][r] = __expf(-0.5f * (accH[t][i] + ch));
      sLL[m][r] = __expf(-0.5f * (accL[t][i] + cl));
    }
  }
  __syncthreads();

  // ---------- Phase 3: Karnik-Mendel scans, split across wave pairs ----------
  const int  s   = tid & 63;       // sample within block
  const float* UU = sUU[s];
  const float* LL = sLL[s];

  if (tid < 64) {
    // KM-left (uses sorted c1) -- waves 0,1
    float s0 = 0.f, t0 = 0.f;
    for (int j = 0; j < 64; ++j) {
      float l = LL[sP1[j]];
      s0 = fmaf(l, sB1[j], s0);
      t0 += l;
    }
    const float sdt = s0 / t0;
    float cs = s0, ct = t0;
    float cmin = __builtin_inff(), best = __builtin_inff();
    int loc = 0;
    for (int j = 0; j < 64; ++j) {
      const int p = sP1[j];
      float d = UU[p] - LL[p];
      cs = fmaf(sB1[j], d, cs);
      ct += d;
      cmin = fminf(cmin, cs / ct);
      float val = fminf(sdt, cmin);
      if (val < best) { best = val; loc = j; }   // jnp.argmin: first occurrence
    }
    float num = 0.f, den = 0.f;
    for (int j = 0; j < 64; ++j) {
      const int p = sP1[j];
      float lu = (j <= loc) ? UU[p] : LL[p];
      num = fmaf(sC1[j], lu, num);               // reference multiplies natural-order c1
      den += lu;
    }
    sYL[s] = num / den;
  } else {
    // KM-right (uses sorted c2) -- waves 2,3
    float s0 = 0.f, t0 = 0.f;
    for (int j = 0; j < 64; ++j) {
      float u = UU[sP2[j]];
      s0 = fmaf(u, sB2[j], s0);
      t0 += u;
    }
    const float sdt = s0 / t0;
    float cs = s0, ct = t0;
    float cmax = -__builtin_inff(), best = -__builtin_inff();
    int loc = 0;
    for (int j = 0; j < 64; ++j) {
      const int p = sP2[j];
      float d = LL[p] - UU[p];
      cs = fmaf(sB2[j], d, cs);
      ct += d;
      cmax = fmaxf(cmax, cs / ct);
      float val = fmaxf(sdt, cmax);
      if (val > best) { best = val; loc = j; }   // jnp.argmax: first occurrence
    }
    float num = 0.f, den = 0.f;
    for (int j = 0; j < 64; ++j) {
      const int p = sP2[j];
      float rl = (j <= loc) ? LL[p] : UU[p];
      num = fmaf(sC2[j], rl, num);
      den += rl;
    }
    const float yR = num / den;
    __syncthreads();                              // wait for sYL
    out[blockBase + s] = 0.5f * (sYL[s] + yR);
    return;
  }
  __syncthreads();                                // matching barrier for waves 0,1
}

extern "C" void kernel_launch(void* const* d_in, const int* in_sizes, int n_in,
                              void* d_out, int out_size, void* d_ws, size_t ws_size,
                              hipStream_t stream) {
  const float* x  = (const float*)d_in[0];
  const float* W  = (const float*)d_in[1];
  const float* c1 = (const float*)d_in[2];
  const float* c2 = (const float*)d_in[3];
  float* ws  = (float*)d_ws;
  float* out = (float*)d_out;

  t2fls_setup<<<1, 64, 0, stream>>>(W, c1, c2, ws);

  const int N = in_sizes[0] / 16;          // 65536 samples
  t2fls_main<<<N / 64, 128, 0, stream>>>(x, c1, c2, ws, out);
}